// MorphologicalFeatureExtractor_49778670961275
// MI455X (gfx1250) — compile-verified
//
#include <hip/hip_runtime.h>
#include <hip/hip_bf16.h>

#define HH 512
#define WW 512
#define HWSZ (HH * WW)
#define NBATCH 32
#define NBLK 32                         // blocks per batch
#define THREADS 256                     // 8 waves (wave32)
#define WAVES (THREADS / 32)
#define PIX_PER_BLK (HWSZ / NBLK)       // 8192
#define F4_PER_THREAD (PIX_PER_BLK / (THREADS * 4))  // 8
#define EPSV 1e-06f
#define WS_STRIDE 12                    // 9 used, padded

typedef float f4  __attribute__((ext_vector_type(4)));
typedef float v2f __attribute__((ext_vector_type(2)));
typedef float v8f __attribute__((ext_vector_type(8)));

// Full 32-lane sum. Stage 16 is done with V_WMMA_F32_16X16X4_F32:
// A = ones(16x4), B vgpr0 = v (B[0][n]=v[n], B[1][n]=v[n+16]), B vgpr1 = 0.
// D[m][n] = sum_k B[k][n] = v[n] + v[n+16]; D vgpr0 holds row m=0 in lanes
// 0-15 and row m=8 (same column values) in lanes 16-31, so every lane gets
// s[lane & 15]. Offsets 8/4/2/1 finish via xor-shuffles.
__device__ __forceinline__ float wave_sum(float v) {
  v2f a;  a[0] = 1.0f; a[1] = 1.0f;
  v2f bb; bb[0] = v;   bb[1] = 0.0f;
  v8f c = {};
  c = __builtin_amdgcn_wmma_f32_16x16x4_f32(false, a, false, bb,
                                            (short)0, c, false, false);
  float s = c[0];
  s += __shfl_xor(s, 8, 32);
  s += __shfl_xor(s, 4, 32);
  s += __shfl_xor(s, 2, 32);
  s += __shfl_xor(s, 1, 32);
  return s;
}

__device__ __forceinline__ int wave_imin(int v) {
  v = min(v, __shfl_xor(v, 16, 32));
  v = min(v, __shfl_xor(v, 8, 32));
  v = min(v, __shfl_xor(v, 4, 32));
  v = min(v, __shfl_xor(v, 2, 32));
  v = min(v, __shfl_xor(v, 1, 32));
  return v;
}

__device__ __forceinline__ int wave_imax(int v) {
  v = max(v, __shfl_xor(v, 16, 32));
  v = max(v, __shfl_xor(v, 8, 32));
  v = max(v, __shfl_xor(v, 4, 32));
  v = max(v, __shfl_xor(v, 2, 32));
  v = max(v, __shfl_xor(v, 1, 32));
  return v;
}

__global__ void __launch_bounds__(THREADS)
morph_reduce(const float* __restrict__ in, float* __restrict__ ws) {
  const int b   = blockIdx.x / NBLK;
  const int blk = blockIdx.x % NBLK;
  const int tid = threadIdx.x;
  const float* base = in + (size_t)b * 3 * HWSZ;

  float areaP = 0.f, areaV = 0.f, areaFG = 0.f, confP = 0.f, confV = 0.f;
  int rmin = HH, rmax = -1, cmin = WW, cmax = -1;

#pragma unroll
  for (int it = 0; it < F4_PER_THREAD; ++it) {
    const int p = blk * PIX_PER_BLK + (it * THREADS + tid) * 4;
    f4 l0 = *(const f4*)(base + p);
    f4 l1 = *(const f4*)(base + HWSZ + p);
    f4 l2 = *(const f4*)(base + 2 * HWSZ + p);
#pragma unroll
    for (int j = 0; j < 4; ++j) {
      float a0 = l0[j], a1 = l1[j], a2 = l2[j];
      // argmax with first-max tie-breaking (matches jnp.argmax)
      int pred = (a1 > a0) ? 1 : 0;
      float mv = fmaxf(a0, a1);
      pred = (a2 > mv) ? 2 : pred;
      float m = fmaxf(mv, a2);
      float e0 = __expf(a0 - m);
      float e1 = __expf(a1 - m);
      float e2 = __expf(a2 - m);
      float inv = 1.0f / (e0 + e1 + e2);
      bool isP = (pred == 1), isV = (pred == 2);
      areaP  += isP ? 1.0f : 0.0f;
      areaV  += isV ? 1.0f : 0.0f;
      areaFG += (pred != 0) ? 1.0f : 0.0f;
      confP  += isP ? e1 * inv : 0.0f;
      confV  += isV ? e2 * inv : 0.0f;
      if (isP) {
        int h = (p + j) >> 9;
        int w = (p + j) & (WW - 1);
        rmin = min(rmin, h); rmax = max(rmax, h);
        cmin = min(cmin, w); cmax = max(cmax, w);
      }
    }
  }

  // convergent point: EXEC all-ones for WMMA reduction
  areaP  = wave_sum(areaP);
  areaV  = wave_sum(areaV);
  areaFG = wave_sum(areaFG);
  confP  = wave_sum(confP);
  confV  = wave_sum(confV);
  rmin = wave_imin(rmin); rmax = wave_imax(rmax);
  cmin = wave_imin(cmin); cmax = wave_imax(cmax);

  __shared__ float lred[WAVES][9];
  const int wave = tid >> 5, lane = tid & 31;
  if (lane == 0) {
    lred[wave][0] = areaP;  lred[wave][1] = areaV; lred[wave][2] = areaFG;
    lred[wave][3] = confP;  lred[wave][4] = confV;
    lred[wave][5] = (float)rmin; lred[wave][6] = (float)rmax;
    lred[wave][7] = (float)cmin; lred[wave][8] = (float)cmax;
  }
  __syncthreads();
  if (tid == 0) {
    float s0 = 0.f, s1 = 0.f, s2 = 0.f, s3 = 0.f, s4 = 0.f;
    float mn5 = (float)HH, mx6 = -1.f, mn7 = (float)WW, mx8 = -1.f;
    for (int k = 0; k < WAVES; ++k) {
      s0 += lred[k][0]; s1 += lred[k][1]; s2 += lred[k][2];
      s3 += lred[k][3]; s4 += lred[k][4];
      mn5 = fminf(mn5, lred[k][5]); mx6 = fmaxf(mx6, lred[k][6]);
      mn7 = fminf(mn7, lred[k][7]); mx8 = fmaxf(mx8, lred[k][8]);
    }
    float* q = ws + (size_t)blockIdx.x * WS_STRIDE;
    q[0] = s0; q[1] = s1; q[2] = s2; q[3] = s3; q[4] = s4;
    q[5] = mn5; q[6] = mx6; q[7] = mn7; q[8] = mx8;
  }
}

__global__ void morph_finalize(const float* __restrict__ ws,
                               float* __restrict__ out) {
  const int b = threadIdx.x;
  if (b >= NBATCH) return;
  float areaP = 0.f, areaV = 0.f, areaFG = 0.f, confP = 0.f, confV = 0.f;
  float rmin = (float)HH, rmax = -1.f, cmin = (float)WW, cmax = -1.f;
  for (int k = 0; k < NBLK; ++k) {
    const float* q = ws + (size_t)(b * NBLK + k) * WS_STRIDE;
    areaP += q[0]; areaV += q[1]; areaFG += q[2];
    confP += q[3]; confV += q[4];
    rmin = fminf(rmin, q[5]); rmax = fmaxf(rmax, q[6]);
    cmin = fminf(cmin, q[7]); cmax = fmaxf(cmax, q[8]);
  }
  const bool ne = areaP > 0.0f;
  const float hr = ne ? (rmax - rmin) : 0.0f;
  const float wr = ne ? (cmax - cmin) : 0.0f;
  const float hw = (float)HWSZ;
  float o[10];
  o[0] = areaP / (areaV + EPSV);
  o[1] = areaP / (areaFG + EPSV);
  o[2] = confP;
  o[3] = hr / (float)HH;
  o[4] = wr / (float)WW;
  o[5] = 2.0f * (hr + wr) / (float)(HH + WW);
  o[6] = confV;
  o[7] = ne ? (confP / (areaP + EPSV)) : 0.0f;
  o[8] = areaFG / hw;
  o[9] = areaP / hw;
  for (int i = 0; i < 10; ++i) out[b * 10 + i] = o[i];
}

extern "C" void kernel_launch(void* const* d_in, const int* in_sizes, int n_in,
                              void* d_out, int out_size, void* d_ws, size_t ws_size,
                              hipStream_t stream) {
  (void)in_sizes; (void)n_in; (void)out_size; (void)ws_size;
  const float* in = (const float*)d_in[0];
  float* out = (float*)d_out;
  float* ws = (float*)d_ws;
  morph_reduce<<<NBATCH * NBLK, THREADS, 0, stream>>>(in, ws);
  morph_finalize<<<1, 32, 0, stream>>>(ws, out);
}